// MessagePassing_12257836663109
// MI455X (gfx1250) — compile-verified
//
#include <hip/hip_runtime.h>
#include <hip/hip_bf16.h>
#include <stdint.h>

// Segment-sum message passing:
//   out[i,:] += X[j,:]  for each edge (i, j)
// N_NODES=100000, N_EDGES=1600000, D_FEAT=32.
// edge_index arrives as int32 pairs (harness: integer -> const int*),
// row-major: edges[2e] = i (dest), edges[2e+1] = j (src).

#define D_FEAT 32
#define THREADS 256
#define WAVES_PER_BLOCK (THREADS / 32)
#define CHUNK 512              // edges staged per LDS chunk (512 * 8B = 4KB)
#define GROUPS (CHUNK / 4)     // 4-edge groups per chunk

#if __has_builtin(__builtin_amdgcn_global_load_async_to_lds_b128)
#define HAVE_ASYNC 1
#else
#define HAVE_ASYNC 0
#endif

#if __has_builtin(__builtin_amdgcn_s_wait_asynccnt)
#define WAIT_ASYNC(n) __builtin_amdgcn_s_wait_asynccnt(n)
#else
#define WAIT_ASYNC(n) asm volatile("s_wait_asynccnt %0" ::"i"(n) : "memory")
#endif

// ---------------------------------------------------------------- zero init
__global__ __launch_bounds__(THREADS) void gnn_zero_kernel(float* __restrict__ out,
                                                           int n4, int n) {
    const int idx = blockIdx.x * blockDim.x + threadIdx.x;
    const int stride = gridDim.x * blockDim.x;
    const float4 z = make_float4(0.f, 0.f, 0.f, 0.f);
    for (int i = idx; i < n4; i += stride) ((float4*)out)[i] = z;
    for (int i = n4 * 4 + idx; i < n; i += stride) out[i] = 0.f;
}

#if HAVE_ASYNC
typedef int v4i_t __attribute__((ext_vector_type(4)));

__device__ __forceinline__ void async_copy16(const long long* gp, long long* lp) {
    // 16B per lane, global -> LDS, tracked by ASYNCcnt (CDNA5 async path).
    // Builtin signature (from hipcc diagnostic): (int4 AS1*, int4 AS3*, Ii, Ii).
    __builtin_amdgcn_global_load_async_to_lds_b128(
        (__attribute__((address_space(1))) v4i_t*)gp,
        (__attribute__((address_space(3))) v4i_t*)lp,
        0, 0);
}
#endif

// ---------------------------------------------------------------- scatter-add
// Layout: one wave per edge, lane = feature (D_FEAT == wave32 width).
__global__ __launch_bounds__(THREADS) void gnn_scatter_kernel(
    const float* __restrict__ X, const int* __restrict__ edges_i32,
    float* __restrict__ out, int n_edges) {
    const int lane = threadIdx.x & 31;
    const int wave = threadIdx.x >> 5;
    const long long* edges = (const long long*)edges_i32;  // one (i,j) pair per edge

#if HAVE_ASYNC
    __shared__ __align__(16) long long lbuf[2][CHUNK];
    const int n_chunks = n_edges / CHUNK;
    const int g = gridDim.x;

    int c = blockIdx.x;
    int buf = 0;
    if (c < n_chunks) {
        async_copy16(edges + (size_t)c * CHUNK + threadIdx.x * 2,
                     &lbuf[0][threadIdx.x * 2]);
    }
    for (; c < n_chunks; c += g, buf ^= 1) {
        const int next = c + g;
        if (next < n_chunks) {
            async_copy16(edges + (size_t)next * CHUNK + threadIdx.x * 2,
                         &lbuf[buf ^ 1][threadIdx.x * 2]);
            WAIT_ASYNC(1);   // current chunk landed; next may stay in flight
        } else {
            WAIT_ASYNC(0);
        }
        __syncthreads();     // all waves' portions of chunk c visible in LDS

        for (int grp = wave; grp < GROUPS; grp += WAVES_PER_BLOCK) {
            const long long* ep4 = &lbuf[buf][grp * 4];
            int ii[4], jj[4];
            float v[4];
#pragma unroll
            for (int k = 0; k < 4; ++k) {
                const long long ep = ep4[k];  // LDS broadcast read
                ii[k] = __builtin_amdgcn_readfirstlane((int)ep);
                jj[k] = __builtin_amdgcn_readfirstlane((int)(ep >> 32));
            }
#pragma unroll
            for (int k = 0; k < 4; ++k)
                v[k] = X[(size_t)jj[k] * D_FEAT + lane];   // 128B coalesced gather
#pragma unroll
            for (int k = 0; k < 4; ++k)
                unsafeAtomicAdd(out + (size_t)ii[k] * D_FEAT + lane, v[k]);
        }
        __syncthreads();     // done reading lbuf[buf] before it is reissued
    }

    // Remainder edges (n_edges % CHUNK), handled by block 0 directly.
    if (blockIdx.x == 0) {
        for (int e = n_chunks * CHUNK + wave; e < n_edges; e += WAVES_PER_BLOCK) {
            const long long ep = edges[e];
            const int i = __builtin_amdgcn_readfirstlane((int)ep);
            const int j = __builtin_amdgcn_readfirstlane((int)(ep >> 32));
            const float v = X[(size_t)j * D_FEAT + lane];
            unsafeAtomicAdd(out + (size_t)i * D_FEAT + lane, v);
        }
    }
#else
    // Fallback: direct streaming of the edge list with hardware prefetch.
    const int gw = blockIdx.x * WAVES_PER_BLOCK + wave;
    const int nw = gridDim.x * WAVES_PER_BLOCK;
    for (int e = gw; e < n_edges; e += nw) {
        __builtin_prefetch(edges + e + 2 * nw, 0, 1);  // global_prefetch_b8
        const long long ep = edges[e];
        const int i = __builtin_amdgcn_readfirstlane((int)ep);
        const int j = __builtin_amdgcn_readfirstlane((int)(ep >> 32));
        const float v = X[(size_t)j * D_FEAT + lane];
        unsafeAtomicAdd(out + (size_t)i * D_FEAT + lane, v);
    }
#endif
}

// ---------------------------------------------------------------- launch
extern "C" void kernel_launch(void* const* d_in, const int* in_sizes, int n_in,
                              void* d_out, int out_size, void* d_ws, size_t ws_size,
                              hipStream_t stream) {
    const float* X = (const float*)d_in[0];
    const int* edges = (const int*)d_in[1];
    float* out = (float*)d_out;

    const int n_edges = in_sizes[1] / 2;
    const int n_out = out_size;
    const int n4 = n_out / 4;

    int zgrid = (n4 + THREADS - 1) / THREADS;
    if (zgrid > 4096) zgrid = 4096;
    if (zgrid < 1) zgrid = 1;
    gnn_zero_kernel<<<zgrid, THREADS, 0, stream>>>(out, n4, n_out);

    const int n_chunks = n_edges / CHUNK;
    int grid = n_chunks;
    if (grid > 1024) grid = 1024;
    if (grid < 1) grid = 1;
    gnn_scatter_kernel<<<grid, THREADS, 0, stream>>>(X, edges, out, n_edges);
}